// LSTMDecoder_51410758533323
// MI455X (gfx1250) — compile-verified
//
#include <hip/hip_runtime.h>
#include <math.h>

// ---------------------------------------------------------------------------
// Problem constants (from reference)
// ---------------------------------------------------------------------------
#define Bsz   32
#define Tlen  64
#define Slen  96
#define Dm    512          // d_model
#define Hm    512          // hidden
#define VOC   32000

typedef float v2f __attribute__((ext_vector_type(2)));
typedef float v8f __attribute__((ext_vector_type(8)));

__device__ __forceinline__ v8f wmma_f32_k4(v2f a, v2f b, v8f c) {
    // D(16x16,f32) += A(16x4,f32) * B(4x16,f32)   [CDNA5 V_WMMA_F32_16X16X4_F32]
    return __builtin_amdgcn_wmma_f32_16x16x4_f32(
        /*neg_a=*/false, a, /*neg_b=*/false, b,
        /*c_mod=*/(short)0, c, /*reuse_a=*/false, /*reuse_b=*/false);
}

__device__ __forceinline__ float sigmoidf_(float x) { return 1.0f / (1.0f + expf(-x)); }

// ---------------------------------------------------------------------------
// Generic K=512 GEMM slab: acc[g] += X[32x512] @ W[g*512 + jtile][wcol0..+511]^T
//   arow : A-matrix row this lane loads (batch index m*16 + (lane&15))
//   jcol : B column this lane loads  (n*16 + (lane&15)), gate g adds g*512
//   hv   : lane>>4  (selects K sub-pair per ISA A/B layout)
// ---------------------------------------------------------------------------
template <int NG>
__device__ __forceinline__ void gemm_part(v8f (&acc)[NG],
                                          const float* __restrict__ X,
                                          const float* __restrict__ W,
                                          int ldw, int wcol0,
                                          int arow, int jcol, int hv) {
    const float* xp = X + (size_t)arow * Dm + hv * 2;
#pragma unroll 4
    for (int k = 0; k < 512; k += 4) {
        v2f a = *(const v2f*)(xp + k);
#pragma unroll
        for (int g = 0; g < NG; ++g) {
            const float* wp = W + (size_t)(g * 512 + jcol) * ldw + wcol0 + k + hv * 2;
            v2f b = *(const v2f*)wp;
            acc[g] = wmma_f32_k4(a, b, acc[g]);
        }
    }
}

// ---------------------------------------------------------------------------
// Zero-init recurrent state (h0 buf0, h1 buf0, c0, c1) each 32x512 floats
// ---------------------------------------------------------------------------
__global__ void init_state_kernel(float* __restrict__ ws) {
    int i = blockIdx.x * blockDim.x + threadIdx.x;     // 0..16383
    if (i < Bsz * Hm) {
        ws[i]               = 0.0f;   // h0 buf0
        ws[32768 + i]       = 0.0f;   // h1 buf0
        ws[65536 + i]       = 0.0f;   // c0
        ws[81920 + i]       = 0.0f;   // c1
    }
}

// ---------------------------------------------------------------------------
// Step part 1: embedding gather, d_i = g + h1@W_d^T + b_d, attention, ctx
// One block per batch row. Small & latency bound -> VALU + LDS.
// ---------------------------------------------------------------------------
__global__ __launch_bounds__(256)
void attn_kernel(const int* __restrict__ ids, int t,
                 const float* __restrict__ emb,
                 const float* __restrict__ W_d, const float* __restrict__ b_d,
                 const float* __restrict__ Z, const float* __restrict__ V,
                 const unsigned char* __restrict__ mask,
                 const float* __restrict__ h1_in,
                 float* __restrict__ gbuf, float* __restrict__ ctxbuf) {
    __shared__ float h1s[Dm];
    __shared__ float dis[Dm];
    __shared__ float sc[Slen];

    const int b = blockIdx.x;
    const int tid = threadIdx.x;

    for (int k = tid; k < Dm; k += 256) h1s[k] = h1_in[b * Dm + k];
    __syncthreads();

    const int id = ids[b * Tlen + t];
    const float* erow = emb + (size_t)id * Dm;

    for (int j = tid; j < Dm; j += 256) {
        float gv = erow[j];
        gbuf[b * Dm + j] = gv;
        float acc = gv + b_d[j];
        const float* wr = W_d + (size_t)j * Hm;
        for (int k = 0; k < Hm; ++k) acc = fmaf(h1s[k], wr[k], acc);
        dis[j] = acc;
    }
    __syncthreads();

    const float scale = 0.044194173824159216f;  // 1/sqrt(512)
    for (int s = tid; s < Slen; s += 256) {
        const float* zr = Z + ((size_t)b * Slen + s) * Dm;
        float acc = 0.0f;
        for (int k = 0; k < Dm; ++k) acc = fmaf(dis[k], zr[k], acc);
        sc[s] = mask[b * Slen + s] ? acc * scale : -INFINITY;
    }
    __syncthreads();

    if (tid == 0) {
        float m = -INFINITY;
        for (int s = 0; s < Slen; ++s) m = fmaxf(m, sc[s]);
        float sum = 0.0f;
        if (m > -INFINITY) {
            for (int s = 0; s < Slen; ++s) { float e = expf(sc[s] - m); sc[s] = e; sum += e; }
        } else {
            for (int s = 0; s < Slen; ++s) sc[s] = 0.0f;
        }
        float inv = (sum > 0.0f) ? 1.0f / sum : 0.0f;
        for (int s = 0; s < Slen; ++s) sc[s] *= inv;
    }
    __syncthreads();

    for (int j = tid; j < Dm; j += 256) {
        float acc = 0.0f;
        for (int s = 0; s < Slen; ++s)
            acc = fmaf(sc[s], V[((size_t)b * Slen + s) * Dm + j], acc);
        ctxbuf[b * Dm + j] = acc;
    }
}

// ---------------------------------------------------------------------------
// LSTM layer 0: gates = [g,ctx]@W_ih0^T + h0@W_hh0^T (+biases), fused cell.
// 64 waves: wave = (m,n), m in {0,1} (batch tile), n in 0..31 (16 units each).
// Four f32-WMMA accumulators (i,f,g,o) -> cell update entirely in-register.
// ---------------------------------------------------------------------------
__global__ __launch_bounds__(256)
void lstm0_kernel(const float* __restrict__ g, const float* __restrict__ ctx,
                  const float* __restrict__ h0_in, float* __restrict__ h0_out,
                  float* __restrict__ c0,
                  const float* __restrict__ W_ih, const float* __restrict__ W_hh,
                  const float* __restrict__ b_ih, const float* __restrict__ b_hh) {
    const int wave = (blockIdx.x * 256 + threadIdx.x) >> 5;  // 0..63
    const int lane = threadIdx.x & 31;
    const int m = wave >> 5, n = wave & 31;
    const int hv = lane >> 4, nn = lane & 15;
    const int arow = m * 16 + nn;
    const int jcol = n * 16 + nn;

    const v8f z = {0.f,0.f,0.f,0.f,0.f,0.f,0.f,0.f};
    v8f acc[4] = {z, z, z, z};

    gemm_part<4>(acc, g,     W_ih, 1024,   0, arow, jcol, hv);   // x[:, :512]
    gemm_part<4>(acc, ctx,   W_ih, 1024, 512, arow, jcol, hv);   // x[:, 512:]
    gemm_part<4>(acc, h0_in, W_hh,  512,   0, arow, jcol, hv);   // h @ W_hh^T

    const float bI = b_ih[jcol]          + b_hh[jcol];
    const float bF = b_ih[512  + jcol]   + b_hh[512  + jcol];
    const float bG = b_ih[1024 + jcol]   + b_hh[1024 + jcol];
    const float bO = b_ih[1536 + jcol]   + b_hh[1536 + jcol];

#pragma unroll
    for (int r = 0; r < 8; ++r) {
        const int br = m * 16 + hv * 8 + r;                      // batch row
        float iv = acc[0][r] + bI;
        float fv = acc[1][r] + bF;
        float gv = acc[2][r] + bG;
        float ov = acc[3][r] + bO;
        float cold = c0[br * Hm + jcol];
        float cn = sigmoidf_(fv) * cold + sigmoidf_(iv) * tanhf(gv);
        c0[br * Hm + jcol] = cn;
        h0_out[br * Hm + jcol] = sigmoidf_(ov) * tanhf(cn);
    }
}

// ---------------------------------------------------------------------------
// LSTM layer 1: gates = h0n@W_ih1^T + h1@W_hh1^T, fused cell.
// ---------------------------------------------------------------------------
__global__ __launch_bounds__(256)
void lstm1_kernel(const float* __restrict__ h0_new, const float* __restrict__ h1_in,
                  float* __restrict__ h1_out, float* __restrict__ c1,
                  const float* __restrict__ W_ih, const float* __restrict__ W_hh,
                  const float* __restrict__ b_ih, const float* __restrict__ b_hh) {
    const int wave = (blockIdx.x * 256 + threadIdx.x) >> 5;
    const int lane = threadIdx.x & 31;
    const int m = wave >> 5, n = wave & 31;
    const int hv = lane >> 4, nn = lane & 15;
    const int arow = m * 16 + nn;
    const int jcol = n * 16 + nn;

    const v8f z = {0.f,0.f,0.f,0.f,0.f,0.f,0.f,0.f};
    v8f acc[4] = {z, z, z, z};

    gemm_part<4>(acc, h0_new, W_ih, 512, 0, arow, jcol, hv);
    gemm_part<4>(acc, h1_in,  W_hh, 512, 0, arow, jcol, hv);

    const float bI = b_ih[jcol]        + b_hh[jcol];
    const float bF = b_ih[512  + jcol] + b_hh[512  + jcol];
    const float bG = b_ih[1024 + jcol] + b_hh[1024 + jcol];
    const float bO = b_ih[1536 + jcol] + b_hh[1536 + jcol];

#pragma unroll
    for (int r = 0; r < 8; ++r) {
        const int br = m * 16 + hv * 8 + r;
        float iv = acc[0][r] + bI;
        float fv = acc[1][r] + bF;
        float gv = acc[2][r] + bG;
        float ov = acc[3][r] + bO;
        float cold = c1[br * Hm + jcol];
        float cn = sigmoidf_(fv) * cold + sigmoidf_(iv) * tanhf(gv);
        c1[br * Hm + jcol] = cn;
        h1_out[br * Hm + jcol] = sigmoidf_(ov) * tanhf(cn);
    }
}

// ---------------------------------------------------------------------------
// o_t = tanh([h1n, ctx] @ W_cat^T + b_cat), stored into O[(b*T + t), :]
// ---------------------------------------------------------------------------
__global__ __launch_bounds__(256)
void ocat_kernel(const float* __restrict__ h1_new, const float* __restrict__ ctx,
                 const float* __restrict__ W_cat, const float* __restrict__ b_cat,
                 int t, float* __restrict__ obuf) {
    const int wave = (blockIdx.x * 256 + threadIdx.x) >> 5;
    const int lane = threadIdx.x & 31;
    const int m = wave >> 5, n = wave & 31;
    const int hv = lane >> 4, nn = lane & 15;
    const int arow = m * 16 + nn;
    const int jcol = n * 16 + nn;

    const v8f z = {0.f,0.f,0.f,0.f,0.f,0.f,0.f,0.f};
    v8f acc[1] = {z};

    gemm_part<1>(acc, h1_new, W_cat, 1024,   0, arow, jcol, hv);
    gemm_part<1>(acc, ctx,    W_cat, 1024, 512, arow, jcol, hv);

    const float bias = b_cat[jcol];
#pragma unroll
    for (int r = 0; r < 8; ++r) {
        const int br = m * 16 + hv * 8 + r;                // batch index
        obuf[((size_t)br * Tlen + t) * Dm + jcol] = tanhf(acc[0][r] + bias);
    }
}

// ---------------------------------------------------------------------------
// Phase 2: logits[2048 x 32000] = O[2048 x 512] @ W_out^T + b_out
// 256,000 waves, one 16x16 tile each, K=512 via 128 f32-WMMA.
// W_out is read exactly once from HBM; O stays in L2.
// ---------------------------------------------------------------------------
__global__ __launch_bounds__(256)
void out_gemm_kernel(const float* __restrict__ o, const float* __restrict__ W_out,
                     const float* __restrict__ b_out, float* __restrict__ logits) {
    const int wave = (blockIdx.x * 256 + threadIdx.x) >> 5;
    const int lane = threadIdx.x & 31;
    const int nt = wave % 2000;          // vocab tile
    const int mt = wave / 2000;          // (b,t) row tile
    const int hv = lane >> 4, nn = lane & 15;

    v8f acc = {0.f,0.f,0.f,0.f,0.f,0.f,0.f,0.f};
    const float* ap = o     + (size_t)(mt * 16 + nn) * Dm + hv * 2;
    const float* bp = W_out + (size_t)(nt * 16 + nn) * Dm + hv * 2;

#pragma unroll 4
    for (int k = 0; k < 512; k += 4) {
        v2f a = *(const v2f*)(ap + k);
        v2f b = *(const v2f*)(bp + k);
        acc = wmma_f32_k4(a, b, acc);
    }

    const float bias = b_out[nt * 16 + nn];
#pragma unroll
    for (int r = 0; r < 8; ++r) {
        const int row = mt * 16 + hv * 8 + r;
        logits[(size_t)row * VOC + nt * 16 + nn] = acc[r] + bias;
    }
}

// ---------------------------------------------------------------------------
// Copy final hidden/cell states into d_out tail: h_n[2,B,H] then c_n[2,B,H]
// ---------------------------------------------------------------------------
__global__ void finalize_kernel(const float* __restrict__ h0, const float* __restrict__ h1,
                                const float* __restrict__ c0, const float* __restrict__ c1,
                                float* __restrict__ tail) {
    int i = blockIdx.x * blockDim.x + threadIdx.x;
    if (i < Bsz * Hm) {
        tail[i]                   = h0[i];
        tail[Bsz * Hm + i]        = h1[i];
        tail[2 * Bsz * Hm + i]    = c0[i];
        tail[3 * Bsz * Hm + i]    = c1[i];
    }
}

// ---------------------------------------------------------------------------
// Host launcher
// ---------------------------------------------------------------------------
extern "C" void kernel_launch(void* const* d_in, const int* in_sizes, int n_in,
                              void* d_out, int out_size, void* d_ws, size_t ws_size,
                              hipStream_t stream) {
    const int*           ids   = (const int*)d_in[0];
    const float*         Z     = (const float*)d_in[1];
    const float*         V     = (const float*)d_in[2];
    const unsigned char* mask  = (const unsigned char*)d_in[3];
    const float*         emb   = (const float*)d_in[4];
    const float*         W_d   = (const float*)d_in[5];
    const float*         b_d   = (const float*)d_in[6];
    const float*         W_ih0 = (const float*)d_in[7];
    const float*         W_hh0 = (const float*)d_in[8];
    const float*         b_ih0 = (const float*)d_in[9];
    const float*         b_hh0 = (const float*)d_in[10];
    const float*         W_ih1 = (const float*)d_in[11];
    const float*         W_hh1 = (const float*)d_in[12];
    const float*         b_ih1 = (const float*)d_in[13];
    const float*         b_hh1 = (const float*)d_in[14];
    const float*         W_cat = (const float*)d_in[15];
    const float*         b_cat = (const float*)d_in[16];
    const float*         W_out = (const float*)d_in[17];
    const float*         b_out = (const float*)d_in[18];

    float* ws = (float*)d_ws;
    float* h0b[2] = {ws,          ws + 16384};   // h0 ping-pong
    float* h1b[2] = {ws + 32768,  ws + 49152};   // h1 ping-pong
    float* c0     = ws + 65536;
    float* c1     = ws + 81920;
    float* gbuf   = ws + 98304;                  // embedded token  [32,512]
    float* ctxbuf = ws + 114688;                 // attention ctx   [32,512]
    float* obuf   = ws + 131072;                 // O matrix [2048,512]

    init_state_kernel<<<64, 256, 0, stream>>>(ws);

    for (int t = 0; t < Tlen; ++t) {
        const float* h0_in = h0b[t & 1];
        float*       h0_out = h0b[(t + 1) & 1];
        const float* h1_in = h1b[t & 1];
        float*       h1_out = h1b[(t + 1) & 1];

        attn_kernel<<<Bsz, 256, 0, stream>>>(ids, t, emb, W_d, b_d, Z, V, mask,
                                             h1_in, gbuf, ctxbuf);
        lstm0_kernel<<<8, 256, 0, stream>>>(gbuf, ctxbuf, h0_in, h0_out, c0,
                                            W_ih0, W_hh0, b_ih0, b_hh0);
        lstm1_kernel<<<8, 256, 0, stream>>>(h0_out, h1_in, h1_out, c1,
                                            W_ih1, W_hh1, b_ih1, b_hh1);
        ocat_kernel<<<8, 256, 0, stream>>>(h1_out, ctxbuf, W_cat, b_cat, t, obuf);
    }

    // 128 row-tiles x 2000 vocab-tiles = 256000 waves = 32000 blocks of 8 waves
    out_gemm_kernel<<<32000, 256, 0, stream>>>(obuf, W_out, b_out, (float*)d_out);

    finalize_kernel<<<64, 256, 0, stream>>>(h0b[0], h1b[0], c0, c1,
                                            (float*)d_out + (size_t)Bsz * Tlen * VOC);
}